// QPSolver_43224550868221
// MI455X (gfx1250) — compile-verified
//
#include <hip/hip_runtime.h>

// Problem constants (from reference)
#define NN     64
#define MM     128
#define BSZ    512
#define NITERS 500
#define ALPHA  1.0f
#define BETA   8.0f

// LDS strides (padded to avoid 64-bank conflicts; all keep 16B alignment)
#define AUG_S 132   // 64 x (64|64) augmented matrix for Gauss-Jordan
#define HS    68    // H rows (128 x 64)
#define PS    132   // PinvHt (64 x 128)
#define DS_   132   // D (128 x 128)
#define PT_S  68    // pinvH transposed (128 x 64)

typedef float v2f __attribute__((ext_vector_type(2)));
typedef float v8f __attribute__((ext_vector_type(8)));

// ---------------------------------------------------------------------------
// CDNA5 f32 WMMA: D(16x16) += A(16x4) * B(4x16)
// A frag: lane l holds A[l&15][2*(l>>4)+v] in a[v]
// B frag: lane l holds B[2*(l>>4)+v][l&15] in b[v]
// C/D   : lane l, reg v holds C[v + 8*(l>>4)][l&15]
// ---------------------------------------------------------------------------
__device__ __forceinline__ v8f wmma_f32_16x16x4(v2f a, v2f b, v8f c) {
  return __builtin_amdgcn_wmma_f32_16x16x4_f32(false, a, false, b, (short)0, c,
                                               false, false);
}

// Tiled GEMM via f32 WMMA. C(16*mtiles x 16*ntiles) = A * B, K = 4*ksteps.
// TA: A element (row,k) read as A[k*lda+row]; TB: B element (k,col) as B[col*ldb+k].
template <bool TA, bool TB>
__device__ __forceinline__ void gemm_wmma(const float* __restrict__ Am, int lda,
                                          const float* __restrict__ Bm, int ldb,
                                          float* __restrict__ Cm, int ldc,
                                          int mtiles, int ntiles, int ksteps,
                                          int wave, int nwaves, int lane) {
  const int l15 = lane & 15;
  const int kh2 = (lane >> 4) * 2;   // 0 or 2
  const int total = mtiles * ntiles;
  for (int t = wave; t < total; t += nwaves) {
    const int mt = t / ntiles, nt = t % ntiles;
    const int arow = mt * 16 + l15;
    const int bcol = nt * 16 + l15;
    v8f acc = {};
    for (int kk = 0; kk < ksteps; ++kk) {
      const int k0 = kk * 4 + kh2;
      v2f a, b;
      a[0] = TA ? Am[k0 * lda + arow] : Am[arow * lda + k0];
      a[1] = TA ? Am[(k0 + 1) * lda + arow] : Am[arow * lda + k0 + 1];
      b[0] = TB ? Bm[bcol * ldb + k0] : Bm[k0 * ldb + bcol];
      b[1] = TB ? Bm[bcol * ldb + k0 + 1] : Bm[(k0 + 1) * ldb + bcol];
      acc = wmma_f32_16x16x4(a, b, acc);
    }
    const int rbase = mt * 16 + (lane >> 4) * 8;
    const int cidx = nt * 16 + l15;
#pragma unroll
    for (int v = 0; v < 8; ++v) Cm[(rbase + v) * ldc + cidx] = acc[v];
  }
}

// Gauss-Jordan inversion of 64x64 in augmented [M | I] LDS buffer (64 x AUG_S).
// 128 threads: thread tid owns column tid of the 128-wide augmented system.
__device__ __forceinline__ void gauss_jordan64(float* aug, float* colbuf,
                                               int tid) {
  for (int k = 0; k < 64; ++k) {
    if (tid < 64) colbuf[tid] = aug[tid * AUG_S + k];
    __syncthreads();
    const float pivinv = 1.0f / colbuf[k];
    aug[k * AUG_S + tid] *= pivinv;
    __syncthreads();
    const float fk = aug[k * AUG_S + tid];  // normalized pivot row, my column
#pragma unroll 4
    for (int i = 0; i < 64; ++i) {
      if (i != k) aug[i * AUG_S + tid] = fmaf(-colbuf[i], fk, aug[i * AUG_S + tid]);
    }
    __syncthreads();
  }
}

// ---------------------------------------------------------------------------
// Kernel 1: per-batch prep.  Produces D (128x128), mu (128), pinvH (64x128).
// ---------------------------------------------------------------------------
__global__ __launch_bounds__(128) void qp_prep(
    const float* __restrict__ qg, const float* __restrict__ bg,
    const float* __restrict__ Pg, const float* __restrict__ Hg,
    float* __restrict__ wsD, float* __restrict__ wsMu,
    float* __restrict__ wsPH) {
  __shared__ __align__(16) float sAug[64 * AUG_S];
  __shared__ __align__(16) float sH[128 * HS];
  __shared__ __align__(16) float sPHt[64 * PS];
  __shared__ float sQ[64], sPq[64], sCol[64];

  const int b = blockIdx.x, tid = threadIdx.x;
  const int lane = tid & 31, wave = tid >> 5;
  const float* Pb = Pg + (size_t)b * (NN * NN);
  const float* Hb = Hg + (size_t)b * (MM * NN);

  // Stage P (augmented with I) and H into LDS
  for (int i = tid; i < NN * NN; i += 128) {
    const int r = i >> 6, c = i & 63;
    sAug[r * AUG_S + c] = Pb[i];
    sAug[r * AUG_S + 64 + c] = (r == c) ? 1.0f : 0.0f;
  }
  for (int i = tid; i < MM * NN; i += 128) {
    const int m = i >> 6, n = i & 63;
    sH[m * HS + n] = Hb[i];
  }
  if (tid < 64) sQ[tid] = qg[(size_t)b * NN + tid];
  __syncthreads();

  // Pinv = P^{-1}  (right half of sAug)
  gauss_jordan64(sAug, sCol, tid);

  // PinvHt(64x128) = Pinv(64x64) * H^T  — WMMA, B read transposed from sH
  gemm_wmma<false, true>(sAug + 64, AUG_S, sH, HS, sPHt, PS, 4, 8, 16, wave, 4,
                         lane);

  // Pinvq (64)
  if (tid < 64) {
    float acc = 0.0f;
    for (int j = 0; j < 64; ++j)
      acc = fmaf(sAug[tid * AUG_S + 64 + j], sQ[j], acc);
    sPq[tid] = acc;
  }
  __syncthreads();  // PinvHt fully written; Pinv reads complete

  // mu = (H * Pinvq - b) / beta
  {
    float acc = 0.0f;
    for (int n = 0; n < 64; ++n) acc = fmaf(sH[tid * HS + n], sPq[n], acc);
    wsMu[(size_t)b * MM + tid] =
        (acc - bg[(size_t)b * MM + tid]) * (1.0f / BETA);
  }

  // D(128x128) = H * PinvHt  — WMMA, written straight to workspace
  gemm_wmma<false, false>(sH, HS, sPHt, PS, wsD + (size_t)b * (MM * MM), MM, 8,
                          8, 16, wave, 4, lane);
  __syncthreads();  // done reading sAug(Pinv); safe to overwrite

  // HtH(64x64) = H^T * H  — WMMA into left half of sAug; identity on right
  gemm_wmma<true, false>(sH, HS, sH, HS, sAug, AUG_S, 4, 4, 32, wave, 4, lane);
  for (int i = tid; i < NN * NN; i += 128) {
    const int r = i >> 6, c = i & 63;
    sAug[r * AUG_S + 64 + c] = (r == c) ? 1.0f : 0.0f;
  }
  __syncthreads();

  gauss_jordan64(sAug, sCol, tid);  // (HtH)^{-1}

  // pinvH(64x128) = (HtH)^{-1} * H^T  — WMMA to workspace
  gemm_wmma<false, true>(sAug + 64, AUG_S, sH, HS, wsPH + (size_t)b * (NN * MM),
                         MM, 4, 8, 16, wave, 4, lane);
}

// ---------------------------------------------------------------------------
// Kernel 2: 500-step fixed-point iteration, D resident in LDS.
//   w = D x_top;  top = (w + x_bot)/beta + mu
//   bot = relu(-(2a/B) w + x_top + (1-2a/B) x_bot - 2a mu)
// ---------------------------------------------------------------------------
__global__ __launch_bounds__(128) void qp_iter(const float* __restrict__ wsD,
                                               const float* __restrict__ wsMu,
                                               float* __restrict__ xsOut) {
  __shared__ __align__(16) float sD[128 * DS_];
  __shared__ __align__(16) float sX[256];

  const int b = blockIdx.x, tid = threadIdx.x;
  const float* Db = wsD + (size_t)b * (MM * MM);
  __builtin_prefetch(Db, 0, 0);  // global_prefetch_b8 warm-up of D stream

  for (int i = tid; i < MM * MM; i += 128) {
    const int r = i >> 7, c = i & 127;
    sD[r * DS_ + c] = Db[i];
  }
  const float mu = wsMu[(size_t)b * MM + tid];
  sX[tid] = 0.0f;
  sX[128 + tid] = 0.0f;

  float* out = xsOut + (size_t)b * (NITERS + 1) * (2 * MM);
  out[tid] = 0.0f;          // X0 top
  out[128 + tid] = 0.0f;    // X0 bot
  __syncthreads();

  const float ib = 1.0f / BETA;            // 0.125
  const float c2 = 2.0f * ALPHA / BETA;    // 0.25
  const float* row = &sD[tid * DS_];

  for (int t = 1; t <= NITERS; ++t) {
    float w = 0.0f;
#pragma unroll 8
    for (int j = 0; j < 128; j += 4) {
      const float4 d = *(const float4*)(row + j);
      const float4 x = *(const float4*)(&sX[j]);
      w = fmaf(d.x, x.x, w);
      w = fmaf(d.y, x.y, w);
      w = fmaf(d.z, x.z, w);
      w = fmaf(d.w, x.w, w);
    }
    const float xt = sX[tid];
    const float xb = sX[128 + tid];
    const float top = fmaf(w + xb, ib, mu);
    float bot = fmaf(-c2, w, xt) + fmaf(1.0f - c2, xb, -2.0f * ALPHA * mu);
    bot = fmaxf(bot, 0.0f);
    __syncthreads();
    sX[tid] = top;
    sX[128 + tid] = bot;
    out[(size_t)t * 256 + tid] = top;
    out[(size_t)t * 256 + 128 + tid] = bot;
    __syncthreads();
  }
}

// ---------------------------------------------------------------------------
// Kernel 3: primal[t][n] = sum_m pinvH[n][m] * (z[t][m] - b[m])  via WMMA.
// Per batch: C(512pad x 64) = Z'(512 x 128) * pinvH^T(128 x 64).
// 256 threads = 8 waves; each wave owns 4 t-tiles with 4 n-tile accumulators.
// ---------------------------------------------------------------------------
__global__ __launch_bounds__(256) void qp_primal(
    const float* __restrict__ wsPH, const float* __restrict__ bg,
    const float* __restrict__ xs, float* __restrict__ primal) {
  __shared__ __align__(16) float sPT[128 * PT_S];  // [m][n] = pinvH[n][m]
  __shared__ float sB[128];

  const int b = blockIdx.x, tid = threadIdx.x;
  const float* ph = wsPH + (size_t)b * (NN * MM);  // row-major n x m
  for (int i = tid; i < NN * MM; i += 256) {
    const int n = i >> 7, m = i & 127;
    sPT[m * PT_S + n] = ph[i];
  }
  if (tid < 128) sB[tid] = bg[(size_t)b * MM + tid];
  __syncthreads();

  const int lane = tid & 31, wave = tid >> 5;
  const int l15 = lane & 15;
  const int kh2 = (lane >> 4) * 2;
  const float* z = xs + (size_t)b * (NITERS + 1) * 256 + 128;  // z[t][m]
  float* outp = primal + (size_t)b * (NITERS + 1) * NN;

  for (int tt = wave; tt < 32; tt += 8) {  // 32 t-tiles of 16 (501 -> pad 512)
    const int trow = tt * 16 + l15;
    const int tr = (trow > NITERS) ? NITERS : trow;  // clamp, no divergence
    const float* zr = z + (size_t)tr * 256;
    v8f acc0 = {}, acc1 = {}, acc2 = {}, acc3 = {};
    for (int kk = 0; kk < 32; ++kk) {
      const int k0 = kk * 4 + kh2;
      v2f a;
      a[0] = zr[k0] - sB[k0];
      a[1] = zr[k0 + 1] - sB[k0 + 1];
      const float* p0 = &sPT[k0 * PT_S];
      const float* p1 = &sPT[(k0 + 1) * PT_S];
      v2f b0 = {p0[l15], p1[l15]};
      v2f b1 = {p0[16 + l15], p1[16 + l15]};
      v2f b2 = {p0[32 + l15], p1[32 + l15]};
      v2f b3 = {p0[48 + l15], p1[48 + l15]};
      acc0 = wmma_f32_16x16x4(a, b0, acc0);
      acc1 = wmma_f32_16x16x4(a, b1, acc1);
      acc2 = wmma_f32_16x16x4(a, b2, acc2);
      acc3 = wmma_f32_16x16x4(a, b3, acc3);
    }
    const int rbase = tt * 16 + (lane >> 4) * 8;
#pragma unroll
    for (int v = 0; v < 8; ++v) {
      const int t = rbase + v;
      if (t <= NITERS) {
        float* o = outp + (size_t)t * NN;
        o[l15] = acc0[v];
        o[16 + l15] = acc1[v];
        o[32 + l15] = acc2[v];
        o[48 + l15] = acc3[v];
      }
    }
  }
}

// ---------------------------------------------------------------------------
extern "C" void kernel_launch(void* const* d_in, const int* in_sizes, int n_in,
                              void* d_out, int out_size, void* d_ws,
                              size_t ws_size, hipStream_t stream) {
  const float* q = (const float*)d_in[0];     // (512, 64)
  const float* bvec = (const float*)d_in[1];  // (512, 128)
  const float* P = (const float*)d_in[2];     // (512, 64, 64)
  const float* H = (const float*)d_in[3];     // (512, 128, 64)
  // d_in[4] = iters (compile-time constant NITERS)

  float* out = (float*)d_out;
  float* xs = out;                                        // (512, 501, 256)
  float* primal = out + (size_t)BSZ * (NITERS + 1) * 256; // (512, 501, 64)

  float* ws = (float*)d_ws;
  float* wsD = ws;                                   // 512 * 16384 floats
  float* wsMu = wsD + (size_t)BSZ * (MM * MM);       // 512 * 128
  float* wsPH = wsMu + (size_t)BSZ * MM;             // 512 * 8192

  qp_prep<<<BSZ, 128, 0, stream>>>(q, bvec, P, H, wsD, wsMu, wsPH);
  qp_iter<<<BSZ, 128, 0, stream>>>(wsD, wsMu, xs);
  qp_primal<<<BSZ, 256, 0, stream>>>(wsPH, bvec, xs, primal);
}